// ToPatches_3513283248782
// MI455X (gfx1250) — compile-verified
//
#include <hip/hip_runtime.h>
#include <stdint.h>

// ---------------------------------------------------------------------------
// ToPatches: x[16,23,120200] f32 -> patches[16,23,1000,200] + mask[16,23,1000]
// Pure data movement (0 FLOPs). ~473 MB HBM traffic -> ~20us at 23.3 TB/s.
// Strategy: stage each 64-patch input span into LDS once (async DMA path),
// then write patches with 128-bit non-temporal stores.
// ---------------------------------------------------------------------------

typedef float vfloat4 __attribute__((ext_vector_type(4)));
typedef float vfloat2 __attribute__((ext_vector_type(2)));
typedef int   v4i     __attribute__((ext_vector_type(4)));

#ifndef __has_builtin
#define __has_builtin(x) 0
#endif

#if __has_builtin(__builtin_amdgcn_global_load_async_to_lds_b128) && \
    __has_builtin(__builtin_amdgcn_s_wait_asynccnt)
#define USE_ASYNC_LDS 1
#else
#define USE_ASYNC_LDS 0
#endif

// Problem constants (match the reference)
constexpr int BB      = 16;
constexpr int CC      = 23;
constexpr int TT      = 120200;
constexpr int PP      = 1000;     // MAX_PATCHES
constexpr int PS      = 200;      // PATCH_SIZE
constexpr int STR     = 150;      // STRIDE
constexpr int ACTUAL  = 801;      // (T - PS)/STR + 1 = 801 < 1000

// Chunking: 64 patches per workgroup. p0 = 64*chunk -> 600*p0 % 16 == 0,
// so every chunk's global span start is 16B aligned.
constexpr int KCH     = 64;
constexpr int NCHUNK  = (ACTUAL + KCH - 1) / KCH;            // 13
constexpr int MAXSPAN_F  = (KCH - 1) * STR + PS;             // 9650 floats
constexpr int MAXSPAN_F4 = (MAXSPAN_F * 4 + 15) / 16 + 2;    // 2415 (padded)

constexpr long long PATCH_OUT_F  = (long long)BB * CC * PP * PS;  // 73,600,000
constexpr int       ZERO_F4_PER_BC = (PP - ACTUAL) * (PS / 4);    // 199*50 = 9950
constexpr int       NZERO_F4     = BB * CC * ZERO_F4_PER_BC;      // 3,661,600
constexpr int       MASK_F4      = BB * CC * PP / 4;              // 92,000

// ---------------------------------------------------------------------------
// Gather kernel: one block per (bc, chunk). Stage span -> LDS, emit patches.
// ---------------------------------------------------------------------------
__global__ __launch_bounds__(256) void ToPatches_gather_kernel(
    const float* __restrict__ x, float* __restrict__ out) {
  __shared__ vfloat4 lds[MAXSPAN_F4];

  const int tid   = threadIdx.x;
  const int chunk = (int)(blockIdx.x % NCHUNK);
  const int bc    = (int)(blockIdx.x / NCHUNK);
  const int p0    = chunk * KCH;
  const int np    = min(KCH, ACTUAL - p0);

  // Input span for this chunk, in floats from start of x.
  const long long gstart    = (long long)bc * TT + (long long)p0 * STR;
  const int       spanF     = (np - 1) * STR + PS;
  const int       spanBytes = spanF * 4;
  const int       nF4       = (spanBytes + 15) >> 4;  // 16B granules to stage

  // ---- Stage global span into LDS ----
#if USE_ASYNC_LDS
  {
    // Builtin signature (from hipcc diagnostic): param0 = v4i AS1* (global,
    // prints as "__device__"), param1 = v4i AS3* (LDS), then imm offset/cpol.
    typedef __attribute__((address_space(1))) v4i* gp_t;
    typedef __attribute__((address_space(3))) v4i* lp_t;
    // Generic->AS casts via integers: a generic global pointer's value is its
    // AS1 address; low 32 bits of a generic LDS pointer are the LDS offset.
    gp_t gsrc = (gp_t)(uintptr_t)(x + gstart);
    lp_t ldst = (lp_t)(uint32_t)(uintptr_t)(&lds[0]);
    for (int i = tid; i < nF4; i += 256) {
      __builtin_amdgcn_global_load_async_to_lds_b128(gsrc + i, ldst + i, 0, 0);
    }
    __builtin_amdgcn_s_wait_asynccnt(0);
  }
  __syncthreads();
#else
  {
    const vfloat4* gsrc = (const vfloat4*)(x + gstart);  // 16B aligned
    for (int i = tid; i < nF4; i += 256) lds[i] = gsrc[i];
  }
  __syncthreads();
#endif

  // ---- Emit patches: np*50 float4 stores, 16B-aligned NT ----
  const float*    ldsF      = (const float*)&lds[0];
  vfloat4*        outv      = (vfloat4*)out;
  const long long outF4Base = ((long long)bc * PP + p0) * (PS / 4);
  const int       nOut      = np * (PS / 4);

  for (int v = tid; v < nOut; v += 256) {
    const int dp = v / (PS / 4);
    const int j4 = v - dp * (PS / 4);
    const int lo = dp * STR + j4 * 4;  // float offset in LDS (8B aligned)
    const vfloat2 a = *(const vfloat2*)(ldsF + lo);
    const vfloat2 b = *(const vfloat2*)(ldsF + lo + 2);
    vfloat4 val;
    val.x = a.x; val.y = a.y; val.z = b.x; val.w = b.y;
    __builtin_nontemporal_store(val, outv + outF4Base + v);
  }
}

// ---------------------------------------------------------------------------
// Epilogue: zero invalid patch slots (p >= 801) and write the mask.
// ---------------------------------------------------------------------------
__global__ __launch_bounds__(256) void ToPatches_epilogue_kernel(
    float* __restrict__ out) {
  const long long i = (long long)blockIdx.x * 256 + threadIdx.x;
  vfloat4* outv = (vfloat4*)out;

  if (i < NZERO_F4) {
    const int bc = (int)(i / ZERO_F4_PER_BC);
    const int r  = (int)(i - (long long)bc * ZERO_F4_PER_BC);
    const long long idx = ((long long)bc * PP + ACTUAL) * (PS / 4) + r;
    vfloat4 z;
    z.x = 0.f; z.y = 0.f; z.z = 0.f; z.w = 0.f;
    __builtin_nontemporal_store(z, outv + idx);
  } else if (i < (long long)NZERO_F4 + MASK_F4) {
    const long long m = i - NZERO_F4;             // f4 index into mask region
    const int r  = (int)(m % (PP / 4));           // f4 within one bc's 1000
    const int p0 = r * 4;
    vfloat4 v;
    v.x = (p0 + 0) < ACTUAL ? 1.f : 0.f;
    v.y = (p0 + 1) < ACTUAL ? 1.f : 0.f;
    v.z = (p0 + 2) < ACTUAL ? 1.f : 0.f;
    v.w = (p0 + 3) < ACTUAL ? 1.f : 0.f;
    __builtin_nontemporal_store(v, outv + (PATCH_OUT_F / 4) + m);
  }
}

// ---------------------------------------------------------------------------
extern "C" void kernel_launch(void* const* d_in, const int* in_sizes, int n_in,
                              void* d_out, int out_size, void* d_ws,
                              size_t ws_size, hipStream_t stream) {
  (void)in_sizes; (void)n_in; (void)out_size; (void)d_ws; (void)ws_size;
  const float* x = (const float*)d_in[0];
  float* out = (float*)d_out;

  const int gatherBlocks = BB * CC * NCHUNK;  // 368 * 13 = 4784
  ToPatches_gather_kernel<<<gatherBlocks, 256, 0, stream>>>(x, out);

  const long long epiElems  = (long long)NZERO_F4 + MASK_F4;  // 3,753,600
  const int       epiBlocks = (int)((epiElems + 255) / 256);  // 14,663
  ToPatches_epilogue_kernel<<<epiBlocks, 256, 0, stream>>>(out);
}